// MyModule_12189117186634
// MI455X (gfx1250) — compile-verified
//
#include <hip/hip_runtime.h>

// CDNA5 / gfx1250: wave32, V_WMMA_F32_16X16X4_F32 (fp32 in / fp32 accum).
//
// The 512 chained Linear(16,16) layers form an affine composition, so we
// fuse groups of G=16 layers on-device (pure reassociation, ~1e-5 relative
// divergence) into 32 effective layers: M_g = W_{g*16+15}...W_{g*16},
// c_g = folded bias. This cuts WMMA issue count and per-layer LDS
// transposes by 16x, pushing the kernel toward the 128 MB / 23.3 TB/s
// (~5.5us) HBM floor. Falls back to the unfused chain if d_ws is too small.

typedef float v2f __attribute__((ext_vector_type(2)));
typedef float v8f __attribute__((ext_vector_type(8)));

#define N_LAYERS 512
#define DIM 16
#define GROUP 16                              // layers fused per group
#define NGROUPS (N_LAYERS / GROUP)            // 32 effective layers
#define TILES 4                               // 16-row tiles per wave
#define WAVES_PER_BLOCK 8
#define LDS_STRIDE 18                         // padded row stride (dwords)
#define ROWS_PER_WAVE (TILES * DIM)           // 64
#define ROWS_PER_BLOCK (WAVES_PER_BLOCK * ROWS_PER_WAVE) // 512

// ---------------------------------------------------------------------------
// Pre-pass: fuse GROUP consecutive layers into one affine map per block.
// Running state: H_out = H_in * M^T + c  with  M <- W_j*M,  c <- W_j*c + b_j.
// Cost is negligible (32 blocks x 15 tiny matmuls).
// ---------------------------------------------------------------------------
__global__ __launch_bounds__(DIM * DIM)
void fuse_groups(const float* __restrict__ W, const float* __restrict__ b,
                 float* __restrict__ Wf, float* __restrict__ bf)
{
    __shared__ float M[DIM * DIM];
    __shared__ float T[DIM * DIM];
    __shared__ float cv[DIM];
    __shared__ float ct[DIM];

    const int g   = blockIdx.x;
    const int tid = threadIdx.x;          // 256 threads: tid = i*16 + k
    const int i   = tid >> 4;
    const int k   = tid & 15;
    const long l0 = (long)g * GROUP;

    M[tid] = W[l0 * DIM * DIM + tid];
    if (tid < DIM) cv[tid] = b[l0 * DIM + tid];
    __syncthreads();

    for (int j = 1; j < GROUP; ++j) {
        const float* __restrict__ Wj = W + (l0 + j) * DIM * DIM;
        float acc = 0.0f;
        #pragma unroll
        for (int t = 0; t < DIM; ++t)
            acc += Wj[i * DIM + t] * M[t * DIM + k];
        T[tid] = acc;
        if (tid < DIM) {
            float a2 = b[(l0 + j) * DIM + tid];
            #pragma unroll
            for (int t = 0; t < DIM; ++t)
                a2 += Wj[tid * DIM + t] * cv[t];
            ct[tid] = a2;
        }
        __syncthreads();
        M[tid] = T[tid];
        if (tid < DIM) cv[tid] = ct[tid];
        __syncthreads();
    }

    Wf[(long)g * DIM * DIM + tid] = M[tid];
    if (tid < DIM) bf[(long)g * DIM + tid] = cv[tid];
}

// ---------------------------------------------------------------------------
// Main chain: state resident in WMMA C/D layout, per-layer D->A transpose
// through wave-private LDS (DS is in-order per wave; wave_barrier only pins
// compiler ordering). nlayers is a runtime arg so fused (32) and unfused
// (512) paths share one kernel.
// ---------------------------------------------------------------------------
__global__ __launch_bounds__(WAVES_PER_BLOCK * 32, 1)
void mlp_chain_wmma(const float* __restrict__ x,
                    const float* __restrict__ W,
                    const float* __restrict__ b,
                    float* __restrict__ out,
                    int nrows, int nlayers)
{
    __shared__ float lds[WAVES_PER_BLOCK][TILES][DIM * LDS_STRIDE];

    const int lane = threadIdx.x & 31;   // wave32
    const int wave = threadIdx.x >> 5;
    const int n    = lane & 15;          // column (B/C/D) / row m (A operand)
    const int h    = lane >> 4;          // lane half

    const long rowBase = (long)blockIdx.x * ROWS_PER_BLOCK + (long)wave * ROWS_PER_WAVE;
    const bool full = (rowBase + ROWS_PER_WAVE) <= (long)nrows;  // wave-uniform

    // ---- load initial state directly into C/D layout -------------------
    v8f acc[TILES];
    if (full) {
        #pragma unroll
        for (int t = 0; t < TILES; ++t)
            #pragma unroll
            for (int r = 0; r < 8; ++r)
                acc[t][r] = x[(rowBase + t * DIM + r + 8 * h) * DIM + n];
    } else {
        #pragma unroll
        for (int t = 0; t < TILES; ++t)
            #pragma unroll
            for (int r = 0; r < 8; ++r) {
                long row = rowBase + t * DIM + r + 8 * h;
                acc[t][r] = (row < nrows) ? x[row * DIM + n] : 0.0f;
            }
    }

    // ---- chained affine layers ----------------------------------------
    for (int layer = 0; layer < nlayers; ++layer) {
        const float* __restrict__ Wl = W + (long)layer * (DIM * DIM);

        // B operand = W^T in 4x16 K-chunks: lane(n,h), comp j <-> K=4kc+2h+j
        v2f bw[4];
        #pragma unroll
        for (int kc = 0; kc < 4; ++kc)
            bw[kc] = *(const v2f*)(Wl + n * DIM + 4 * kc + 2 * h);

        // one shared bias vector in C/D layout: D[M][N] += b[N] = b[lane%16]
        const float bias = b[(long)layer * DIM + n];
        v8f bias_vec;
        #pragma unroll
        for (int i = 0; i < 8; ++i) bias_vec[i] = bias;

        // stage all tiles to LDS (D layout -> row-major tile), wave-private
        #pragma unroll
        for (int t = 0; t < TILES; ++t) {
            float* Ht = &lds[wave][t][0];
            #pragma unroll
            for (int r = 0; r < 8; ++r)
                Ht[(r + 8 * h) * LDS_STRIDE + n] = acc[t][r];
        }
        __builtin_amdgcn_wave_barrier();

        // A fragments: lane(m=n, h), chunk kc -> H[m][4kc+2h .. +1]
        v2f af[TILES][4];
        #pragma unroll
        for (int t = 0; t < TILES; ++t) {
            const float* Ht = &lds[wave][t][0];
            #pragma unroll
            for (int kc = 0; kc < 4; ++kc)
                af[t][kc] = *(const v2f*)(Ht + n * LDS_STRIDE + 4 * kc + 2 * h);
        }
        __builtin_amdgcn_wave_barrier();

        // kc = 0: C source is the shared bias vector (SRC2 != VDST),
        // so no per-tile splat is needed; then chain kc = 1..3 in place.
        // 4 independent accumulation chains interleave (tile-inner).
        v8f c[TILES];
        #pragma unroll
        for (int t = 0; t < TILES; ++t)
            c[t] = __builtin_amdgcn_wmma_f32_16x16x4_f32(
                false, af[t][0], false, bw[0],
                (short)0, bias_vec, false, false);

        #pragma unroll
        for (int kc = 1; kc < 4; ++kc)
            #pragma unroll
            for (int t = 0; t < TILES; ++t)
                c[t] = __builtin_amdgcn_wmma_f32_16x16x4_f32(
                    false, af[t][kc], false, bw[kc],
                    (short)0, c[t], false, false);

        #pragma unroll
        for (int t = 0; t < TILES; ++t) acc[t] = c[t];
    }

    // ---- store final state from C/D layout ----------------------------
    if (full) {
        #pragma unroll
        for (int t = 0; t < TILES; ++t)
            #pragma unroll
            for (int r = 0; r < 8; ++r)
                out[(rowBase + t * DIM + r + 8 * h) * DIM + n] = acc[t][r];
    } else {
        #pragma unroll
        for (int t = 0; t < TILES; ++t)
            #pragma unroll
            for (int r = 0; r < 8; ++r) {
                long row = rowBase + t * DIM + r + 8 * h;
                if (row < nrows) out[row * DIM + n] = acc[t][r];
            }
    }
}

extern "C" void kernel_launch(void* const* d_in, const int* in_sizes, int n_in,
                              void* d_out, int out_size, void* d_ws, size_t ws_size,
                              hipStream_t stream) {
    (void)n_in; (void)out_size;
    const float* x = (const float*)d_in[0];
    const float* W = (const float*)d_in[1];
    const float* b = (const float*)d_in[2];
    float* out = (float*)d_out;

    const int nrows   = in_sizes[0] / DIM;
    const int nblocks = (nrows + ROWS_PER_BLOCK - 1) / ROWS_PER_BLOCK;

    const size_t needWs = (size_t)NGROUPS * (DIM * DIM + DIM) * sizeof(float);
    if (ws_size >= needWs) {
        float* Wf = (float*)d_ws;                       // 32 fused 16x16 weights
        float* bf = Wf + (size_t)NGROUPS * DIM * DIM;   // 32 fused biases
        fuse_groups<<<NGROUPS, DIM * DIM, 0, stream>>>(W, b, Wf, bf);
        mlp_chain_wmma<<<nblocks, WAVES_PER_BLOCK * 32, 0, stream>>>(
            x, Wf, bf, out, nrows, NGROUPS);
    } else {
        // scratch too small: run the exact unfused 512-layer chain
        mlp_chain_wmma<<<nblocks, WAVES_PER_BLOCK * 32, 0, stream>>>(
            x, W, b, out, nrows, N_LAYERS);
    }
}